// GATLayer_66271345377740
// MI455X (gfx1250) — compile-verified
//
#include <hip/hip_runtime.h>

#define NNODES   50000
#define FEAT     128          // HEADS*OUT == IN_DIM == 128
#define HEADS    4
#define NEDGE    800000
#define ETOT     (NEDGE + NNODES)   // self loops appended
#define NEG_SLOPE 0.2f

typedef __attribute__((ext_vector_type(2))) float v2f;
typedef __attribute__((ext_vector_type(8))) float v8f;

// ---------------------------------------------------------------------------
// Kernel 0: init  out[n][c] = bias[c],  m = -inf, s = 0
// ---------------------------------------------------------------------------
__global__ __launch_bounds__(256) void gat_init(float* __restrict__ out,
                                                float* __restrict__ m,
                                                float* __restrict__ s,
                                                const float* __restrict__ bias) {
    int tid = blockIdx.x * 256 + threadIdx.x;
    int stride = gridDim.x * 256;
    for (int i = tid; i < NNODES * FEAT; i += stride)
        out[i] = bias[i & (FEAT - 1)];
    for (int i = tid; i < NNODES * HEADS; i += stride) {
        m[i] = -1e30f;
        s[i] = 0.0f;
    }
}

// ---------------------------------------------------------------------------
// Kernel 1: dual GEMM  xl = x @ W_l, xr = x @ W_r  via v_wmma_f32_16x16x4_f32
// Each wave computes one 16x16 output tile; 8 waves/block span the 128 cols;
// blockIdx.x spans 3125 row tiles (50000/16 exact); blockIdx.z picks W_l/W_r.
// f32 A 16x4 layout: lanes 0-15 -> K=k,k+1 ; lanes 16-31 -> K=k+2,k+3 (M=lane&15)
// B 4x16: v0 = rows k / k+2 across lane halves, v1 = rows k+1 / k+3 (N=lane&15)
// C/D: VGPR r holds rows r (lanes 0-15) and r+8 (lanes 16-31), N=lane&15
// ---------------------------------------------------------------------------
__global__ __launch_bounds__(256) void gat_gemm(const float* __restrict__ x,
                                                const float* __restrict__ Wl,
                                                const float* __restrict__ Wr,
                                                float* __restrict__ xl,
                                                float* __restrict__ xr) {
    const int lane  = threadIdx.x & 31;
    const int wave  = threadIdx.x >> 5;          // 0..7 -> column tile
    const int m0    = blockIdx.x * 16;
    const int n0    = wave * 16;
    const int lhalf = lane >> 4;                 // 0 or 1
    const int l15   = lane & 15;

    const float* __restrict__ W    = blockIdx.z ? Wr : Wl;
    float*       __restrict__ outp = blockIdx.z ? xr : xl;

    const float* __restrict__ arow = x + (m0 + l15) * FEAT;

    v8f c = {0.f, 0.f, 0.f, 0.f, 0.f, 0.f, 0.f, 0.f};

    #pragma unroll 4
    for (int k = 0; k < FEAT; k += 4) {
        const int ka = k + lhalf * 2;
        v2f a, b;
        a.x = arow[ka];
        a.y = arow[ka + 1];
        b.x = W[ka * FEAT + n0 + l15];
        b.y = W[(ka + 1) * FEAT + n0 + l15];
        c = __builtin_amdgcn_wmma_f32_16x16x4_f32(
                /*neg_a=*/false, a, /*neg_b=*/false, b,
                /*c_mod=*/(short)0, c, /*reuse_a=*/false, /*reuse_b=*/false);
    }

    #pragma unroll
    for (int r = 0; r < 8; ++r)
        outp[(m0 + r + lhalf * 8) * FEAT + n0 + l15] = c[r];
}

// ---------------------------------------------------------------------------
// Kernel 2: per-edge scores.  One wave per edge; lane owns 4 channels (float4).
// score[e][h] = sum_c att[h][c] * leakyrelu(xl[src][h][c] + xr[dst][h][c])
// Head h owns lanes 8h..8h+7 -> shfl_xor(1,2,4) reduction stays in-group.
// Atomic float max into m[dst][h] (global_atomic_max_num_f32).
// ---------------------------------------------------------------------------
__global__ __launch_bounds__(256) void gat_score(const float* __restrict__ xl,
                                                 const float* __restrict__ xr,
                                                 const int*   __restrict__ ei,
                                                 const float* __restrict__ att,
                                                 float* __restrict__ score,
                                                 float* __restrict__ m) {
    const int lane = threadIdx.x & 31;
    const int gw   = (blockIdx.x * blockDim.x + threadIdx.x) >> 5;
    const int nw   = (gridDim.x * blockDim.x) >> 5;

    const float4 a = ((const float4*)att)[lane];

    for (int e = gw; e < ETOT; e += nw) {
        int src, dst;
        if (e < NEDGE) { src = ei[e]; dst = ei[NEDGE + e]; }
        else           { src = e - NEDGE; dst = src; }

        const float4 vl = ((const float4*)(xl + src * FEAT))[lane];
        const float4 vr = ((const float4*)(xr + dst * FEAT))[lane];

        float p = 0.f, t;
        t = vl.x + vr.x; t = (t >= 0.f) ? t : NEG_SLOPE * t; p += a.x * t;
        t = vl.y + vr.y; t = (t >= 0.f) ? t : NEG_SLOPE * t; p += a.y * t;
        t = vl.z + vr.z; t = (t >= 0.f) ? t : NEG_SLOPE * t; p += a.z * t;
        t = vl.w + vr.w; t = (t >= 0.f) ? t : NEG_SLOPE * t; p += a.w * t;

        p += __shfl_xor(p, 1, 32);
        p += __shfl_xor(p, 2, 32);
        p += __shfl_xor(p, 4, 32);

        if ((lane & 7) == 0) {
            const int h = lane >> 3;
            score[e * HEADS + h] = p;
            __hip_atomic_fetch_max(&m[dst * HEADS + h], p,
                                   __ATOMIC_RELAXED, __HIP_MEMORY_SCOPE_AGENT);
        }
    }
}

// ---------------------------------------------------------------------------
// Kernel 3: ex = exp(score - m[dst]);  s[dst] += ex  (thread per edge*head)
// ---------------------------------------------------------------------------
__global__ __launch_bounds__(256) void gat_exp(const int*   __restrict__ ei,
                                               float* __restrict__ score,
                                               const float* __restrict__ m,
                                               float* __restrict__ s) {
    const int tot    = ETOT * HEADS;
    const int stride = gridDim.x * blockDim.x;
    for (int i = blockIdx.x * blockDim.x + threadIdx.x; i < tot; i += stride) {
        const int e = i >> 2;
        const int h = i & 3;
        const int dst = (e < NEDGE) ? ei[NEDGE + e] : (e - NEDGE);
        const float ex = __expf(score[i] - m[dst * HEADS + h]);
        score[i] = ex;                       // overwrite with unnormalized alpha
        unsafeAtomicAdd(&s[dst * HEADS + h], ex);
    }
}

// ---------------------------------------------------------------------------
// Kernel 4: out[dst] += (ex/s[dst]) * xl[src].  Wave per edge, float4 per lane,
// 4x global_atomic_add_f32 per lane (L2-resident, out = 25.6 MB).
// ---------------------------------------------------------------------------
__global__ __launch_bounds__(256) void gat_scatter(const float* __restrict__ xl,
                                                   const int*   __restrict__ ei,
                                                   const float* __restrict__ score,
                                                   const float* __restrict__ s,
                                                   float* __restrict__ out) {
    const int lane = threadIdx.x & 31;
    const int h    = lane >> 3;
    const int gw   = (blockIdx.x * blockDim.x + threadIdx.x) >> 5;
    const int nw   = (gridDim.x * blockDim.x) >> 5;

    for (int e = gw; e < ETOT; e += nw) {
        int src, dst;
        if (e < NEDGE) { src = ei[e]; dst = ei[NEDGE + e]; }
        else           { src = e - NEDGE; dst = src; }

        const float alpha = score[e * HEADS + h] / s[dst * HEADS + h];
        const float4 v = ((const float4*)(xl + src * FEAT))[lane];

        float* op = out + dst * FEAT + lane * 4;
        unsafeAtomicAdd(op + 0, alpha * v.x);
        unsafeAtomicAdd(op + 1, alpha * v.y);
        unsafeAtomicAdd(op + 2, alpha * v.z);
        unsafeAtomicAdd(op + 3, alpha * v.w);
    }
}

// ---------------------------------------------------------------------------
extern "C" void kernel_launch(void* const* d_in, const int* in_sizes, int n_in,
                              void* d_out, int out_size, void* d_ws, size_t ws_size,
                              hipStream_t stream) {
    const float* x    = (const float*)d_in[0];   // [N,128]
    const int*   ei   = (const int*)  d_in[1];   // [2,E] flat (row0=src, row1=dst)
    const float* W_l  = (const float*)d_in[2];   // [128,128]
    const float* W_r  = (const float*)d_in[3];   // [128,128]
    const float* att  = (const float*)d_in[4];   // [4,32] flat 128
    const float* bias = (const float*)d_in[5];   // [128]
    float* out = (float*)d_out;                  // [N,128]

    // workspace layout (floats)
    float* ws    = (float*)d_ws;
    float* xl    = ws;                                   // N*128
    float* xr    = xl + NNODES * FEAT;                   // N*128
    float* mbuf  = xr + NNODES * FEAT;                   // N*4
    float* sbuf  = mbuf + NNODES * HEADS;                // N*4
    float* score = sbuf + NNODES * HEADS;                // ETOT*4

    // 0) init out=bias, m=-inf, s=0
    gat_init<<<4096, 256, 0, stream>>>(out, mbuf, sbuf, bias);

    // 1) xl = x@W_l, xr = x@W_r  (WMMA f32; z selects the weight matrix)
    gat_gemm<<<dim3(NNODES / 16, 1, 2), 256, 0, stream>>>(x, W_l, W_r, xl, xr);

    // 2) edge scores + segment max
    gat_score<<<4096, 256, 0, stream>>>(xl, xr, ei, att, score, mbuf);

    // 3) exp + segment sum
    gat_exp<<<8192, 256, 0, stream>>>(ei, score, mbuf, sbuf);

    // 4) normalized weighted scatter-add
    gat_scatter<<<8192, 256, 0, stream>>>(xl, ei, score, sbuf, out);
}